// EventCameraSim_11278584119656
// MI455X (gfx1250) — compile-verified
//
#include <hip/hip_runtime.h>
#include <stdint.h>

// ---- CDNA5 vector types for builtins -------------------------------------
typedef float        v4f __attribute__((ext_vector_type(4)));
typedef unsigned int v4u __attribute__((ext_vector_type(4)));
typedef int          v4i __attribute__((ext_vector_type(4)));
typedef int          v8i __attribute__((ext_vector_type(8)));

#define EPS_LOG 1e-3f
#define THRESH_C 0.15f
#define BLK 128   // 4 wave32s per block

// Byte offset of a generic (LDS-aperture) pointer within the workgroup LDS
// allocation: on gfx1250 the flat LDS address is {aperture_hi, offset[31:0]}.
__device__ __forceinline__ uint32_t lds_byte_offset(const void* p) {
  return (uint32_t)(uintptr_t)p;
}

// Issue one TDM 1-D tile load: nelem floats from gptr -> LDS at lds_off.
// D# layout per CDNA5 ISA ch.8 (CDNA5 ISA). Tracked by TENSORcnt.
__device__ __forceinline__ void tdm_load_1d_f32(uint32_t lds_off,
                                                const void* gptr,
                                                uint32_t nelem) {
  uint64_t ga = (uint64_t)(uintptr_t)gptr;

  v4u g0;
  g0.x = 0x1u;                                   // count=1 (valid), user mode
  g0.y = lds_off;                                // lds_addr (bytes)
  g0.z = (uint32_t)(ga & 0xFFFFFFFFu);           // global_addr[31:0]
  g0.w = (uint32_t)((ga >> 32) & 0x01FFFFFFu)    // global_addr[56:32]
         | 0x80000000u;                          // type = 2 ("image")

  v8i g1;
  g1[0] = 0x00020000;                            // data_size=2 -> 4 bytes
  g1[1] = (int)((nelem & 0xFFFFu) << 16);        // tensor_dim0[15:0]
  g1[2] = (int)(((nelem >> 16) & 0xFFFFu)        // tensor_dim0[31:16]
         | (1u << 16));                          // tensor_dim1 = 1
  g1[3] = (int)((uint32_t)BLK << 16);            // tile_dim0 = 128
  g1[4] = 0;                                     // tile_dim1=0, tile_dim2=0
  g1[5] = BLK;                                   // tensor_dim0_stride = 128
  g1[6] = 0;
  g1[7] = 0;

  v4i z4 = {0, 0, 0, 0};
#if defined(__clang_major__) && (__clang_major__ >= 23)
  v8i z8 = {0, 0, 0, 0, 0, 0, 0, 0};
  __builtin_amdgcn_tensor_load_to_lds(g0, g1, z4, z4, z8, 0);
#else
  __builtin_amdgcn_tensor_load_to_lds(g0, g1, z4, z4, 0);
#endif
}

__global__ void EventCameraSim_kernel(const float* __restrict__ x,
                                      const float* __restrict__ init_img,
                                      const int* __restrict__ time_p,
                                      float* __restrict__ out_t,   // [P,K]
                                      float* __restrict__ out_pol, // [P]
                                      int P, int K) {
  __shared__ float s_x[BLK];
  __shared__ float s_i[BLK];
  __shared__ float s_r[BLK];
  __shared__ float s_It[BLK];
  __shared__ float s_xl[BLK];
  __shared__ float s_pol[BLK];

  const int t    = (int)threadIdx.x;
  const int base = (int)blockIdx.x * BLK;
  const int rem  = (P - base < BLK) ? (P - base) : BLK;

  // ---- Phase 0: wave 0 DMAs both input tiles into LDS via the TDM --------
  if (t < 32) {  // wave-uniform branch (wave32)
    tdm_load_1d_f32(lds_byte_offset(s_x), x + base, (uint32_t)rem);
    tdm_load_1d_f32(lds_byte_offset(s_i), init_img + base, (uint32_t)rem);
    __builtin_amdgcn_s_wait_tensorcnt(0);
  }
  __syncthreads();

  const float time_f  = (float)(*time_p);
  const float delta_t = time_f;  // INITIAL_TIME == 0

  // ---- Phase 1: per-pixel parameters into LDS ----------------------------
  if (t < rem) {
    float xv = s_x[t];
    float iv = s_i[t];
    float It = logf(iv + EPS_LOG);
    float xl = logf(xv + EPS_LOG);
    float dI = xl - It;
    float n_events = floorf(fabsf(dI / THRESH_C));
    float sg  = (dI > 0.0f) ? 1.0f : ((dI < 0.0f) ? -1.0f : 0.0f);
    float pol = (n_events > 0.0f) ? sg : 0.0f;
    float slope = dI / delta_t;
    float slope_safe = (fabsf(pol) > 0.0f) ? slope : 1.0f;
    float r = (pol * THRESH_C) / slope_safe;  // t_k = k*r + time

    s_r[t]   = r;
    s_It[t]  = It;
    s_xl[t]  = xl;
    s_pol[t] = pol;
    __builtin_nontemporal_store(pol, out_pol + base + t);
  }
  __syncthreads();

  const float qnan = __uint_as_float(0x7FC00000u);

  // ---- Phase 2: coalesced streaming (NT) writes of time_events -----------
  if ((K & 3) == 0) {
    // vec4 path: K divisible by 4 -> each float4 stays inside one pixel row.
    const int KV = K >> 2;                       // float4s per pixel
    const long long vbase = (long long)blockIdx.x * BLK * KV;
    for (int it = 0; it < KV; ++it) {
      int v   = it * BLK + t;                    // consecutive lanes -> consecutive vec4s
      int pix = v / KV;
      int kb  = (v - pix * KV) * 4 + 1;
      if (pix < rem) {
        float r = s_r[pix], It = s_It[pix], xl = s_xl[pix], pol = s_pol[pix];
        v4f o;
#pragma unroll
        for (int j = 0; j < 4; ++j) {
          float kf  = (float)(kb + j);
          float lvl = pol * kf * THRESH_C;
          float tv  = fmaf(kf, r, time_f);
          float la  = lvl + It;
          bool valid = ((pol > 0.0f) && (la < xl)) ||
                       ((pol < 0.0f) && (la > xl));
          o[j] = valid ? tv : qnan;
        }
        __builtin_nontemporal_store(o, (v4f*)out_t + (vbase + v));
      }
    }
  } else {
    // scalar fallback for K not a multiple of 4 (still fully coalesced b32 stores)
    const long long ebase = (long long)blockIdx.x * BLK * K;
    const int total = rem * K;
    for (int e = t; e < total; e += BLK) {
      int pix = e / K;
      int k   = e - pix * K + 1;
      float r = s_r[pix], It = s_It[pix], xl = s_xl[pix], pol = s_pol[pix];
      float kf  = (float)k;
      float lvl = pol * kf * THRESH_C;
      float tv  = fmaf(kf, r, time_f);
      float la  = lvl + It;
      bool valid = ((pol > 0.0f) && (la < xl)) || ((pol < 0.0f) && (la > xl));
      __builtin_nontemporal_store(valid ? tv : qnan, out_t + ebase + e);
    }
  }
}

extern "C" void kernel_launch(void* const* d_in, const int* in_sizes, int n_in,
                              void* d_out, int out_size, void* d_ws, size_t ws_size,
                              hipStream_t stream) {
  const float* x        = (const float*)d_in[0];
  const float* init_img = (const float*)d_in[1];
  const int*   time_p   = (const int*)d_in[2];
  // d_in[3] (k_max) is recoverable from sizes: out = P*K (time_events) + P (pol)
  const int P = in_sizes[0];
  const int K = out_size / P - 1;

  float* out_t   = (float*)d_out;
  float* out_pol = (float*)d_out + (size_t)P * (size_t)K;

  const int grid = (P + BLK - 1) / BLK;
  hipLaunchKernelGGL(EventCameraSim_kernel, dim3(grid), dim3(BLK), 0, stream,
                     x, init_img, time_p, out_t, out_pol, P, K);
}